// Reweighting_Revision_Loss_v2_39883066311038
// MI455X (gfx1250) — compile-verified
//
#include <hip/hip_runtime.h>
#include <cstdint>

#define ALPHA 0.01f
#define EPSF  1e-15f

// ---------------------------------------------------------------------------
// CDNA5 async global->LDS copy (ASYNCcnt-tracked). Inline asm per
// CDNA5 ISA / CDNA5 ISA: per-lane
//   16 bytes from global address vaddr are written to LDS at offset vdst
// vdst holds the wave-relative LDS byte offset (low 32 bits of the flat
// shared-aperture address), vaddr holds the 64-bit global address.
// ---------------------------------------------------------------------------
__device__ __forceinline__ void async_copy_b128(const float* gsrc, float* ldst) {
    uint32_t lds_off = (uint32_t)(uintptr_t)ldst;   // addr[31:0] == LDS offset
    asm volatile("global_load_async_to_lds_b128 %0, %1, off"
                 :
                 : "v"(lds_off), "v"(gsrc)
                 : "memory");
}

__device__ __forceinline__ void wait_async_zero() {
    asm volatile("s_wait_asynccnt 0" ::: "memory");
}

// ---------------------------------------------------------------------------
// Kernel 1: Tt[t*C + c] = T[c*C + t] + ALPHA * corr[c*C + t]   (transposed)
// so that "column t of T_result" is a contiguous row of Tt.
// ---------------------------------------------------------------------------
__global__ void build_Tt_kernel(const float* __restrict__ T,
                                const float* __restrict__ corr,
                                float* __restrict__ Tt, int C) {
    int o = blockIdx.x * blockDim.x + threadIdx.x;
    int total = C * C;
    if (o >= total) return;
    int t = o / C;
    int c = o - t * C;
    int in = c * C + t;
    Tt[o] = fmaf(ALPHA, corr[in], T[in]);
}

// ---------------------------------------------------------------------------
// Kernel 2: one block (128 threads = 4 wave32) per row.
//   - async-stage Tt[target[i]] row (4KB) into LDS
//   - single-pass online softmax carrying (max m, sum s=Σe, dot d=Σe*w)
//   - deterministic LDS tree merge -> per-row  beta_i * ce_i
// ---------------------------------------------------------------------------
__global__ __launch_bounds__(128) void row_loss_kernel(
    const float* __restrict__ out, const float* __restrict__ Tt,
    const int* __restrict__ tgt, float* __restrict__ rowval, int C) {
    extern __shared__ float smem[];          // [C] tile | [128] m | [128] s | [128] d
    float* tile = smem;
    float* mred = smem + C;
    float* sred = mred + 128;
    float* dred = sred + 128;

    const int i   = blockIdx.x;
    const int tid = threadIdx.x;
    const int t   = tgt[i];
    const float* __restrict__ trow = Tt + (size_t)t * C;
    const float* __restrict__ xrow = out + (size_t)i * C;

    // --- async global->LDS stage of the gathered Tt row (16B per lane) ---
    const int nvec = C >> 2;                              // full float4 chunks
    for (int k = tid; k < nvec; k += blockDim.x)
        async_copy_b128(trow + 4 * k, tile + 4 * k);
    for (int c = (nvec << 2) + tid; c < C; c += blockDim.x)  // tail (C % 4)
        tile[c] = trow[c];
    wait_async_zero();          // this wave's async copies landed in LDS
    __syncthreads();            // all waves' copies landed

    // --- single pass over the row: online softmax + weighted dot ---
    float m = -INFINITY, s = 0.0f, d = 0.0f;
    for (int c = tid; c < C; c += 128) {
        float x = xrow[c];
        float w = tile[c];
        if (x > m) {
            float r = __expf(m - x);     // exp(-inf)=0 handles first element
            s = s * r + 1.0f;
            d = d * r + w;
            m = x;
        } else {
            float e = __expf(x - m);
            s += e;
            d = fmaf(e, w, d);
        }
    }
    mred[tid] = m; sred[tid] = s; dred[tid] = d;
    __syncthreads();

    // --- deterministic tree merge of (m, s, d) triples across 128 threads ---
    for (int off = 64; off > 0; off >>= 1) {
        if (tid < off) {
            float m1 = mred[tid], m2 = mred[tid + off];
            float M  = fmaxf(m1, m2);
            float r1 = __expf(m1 - M), r2 = __expf(m2 - M);
            mred[tid] = M;
            sred[tid] = sred[tid] * r1 + sred[tid + off] * r2;
            dred[tid] = dred[tid] * r1 + dred[tid + off] * r2;
        }
        __syncthreads();
    }

    if (tid == 0) {
        float M = mred[0];          // row max
        float S = sred[0];          // Σ exp(x - M)
        float D = dred[0];          // Σ exp(x - M) * T_result[c, t]
        float xt   = xrow[t];
        float et   = __expf(xt - M);
        float pro1 = et / S;        // p[i, t]
        float pro2 = D / S;         // (p @ T_result)[i, t]
        float beta = pro1 / (pro2 + EPSF);
        float ce   = logf(S) + M - xt;   // -log_softmax[t]
        rowval[i]  = beta * ce;
    }
}

// ---------------------------------------------------------------------------
// Kernel 3: deterministic fixed-order mean over B per-row values.
// ---------------------------------------------------------------------------
__global__ __launch_bounds__(256) void reduce_mean_kernel(
    const float* __restrict__ vals, int B, float* __restrict__ loss) {
    __shared__ float red[256];
    int tid = threadIdx.x;
    float acc = 0.0f;
    for (int i = tid; i < B; i += 256) acc += vals[i];   // fixed order
    red[tid] = acc;
    __syncthreads();
    for (int off = 128; off > 0; off >>= 1) {
        if (tid < off) red[tid] += red[tid + off];
        __syncthreads();
    }
    if (tid == 0) loss[0] = red[0] / (float)B;
}

// ---------------------------------------------------------------------------
extern "C" void kernel_launch(void* const* d_in, const int* in_sizes, int n_in,
                              void* d_out, int out_size, void* d_ws, size_t ws_size,
                              hipStream_t stream) {
    const float* out_logits = (const float*)d_in[0];   // [B, C]
    const float* correction = (const float*)d_in[1];   // [C, C]
    const int*   target     = (const int*)d_in[2];     // [B]
    const float* T          = (const float*)d_in[3];   // [C, C]

    const int B = in_sizes[2];
    const int C = in_sizes[0] / B;

    float* Tt     = (float*)d_ws;                      // C*C floats
    float* rowval = Tt + (size_t)C * C;                // B floats
    float* loss   = (float*)d_out;

    const int totalT = C * C;
    build_Tt_kernel<<<(totalT + 255) / 256, 256, 0, stream>>>(T, correction, Tt, C);

    const size_t smem_bytes = ((size_t)C + 3 * 128) * sizeof(float);
    row_loss_kernel<<<B, 128, smem_bytes, stream>>>(out_logits, Tt, target, rowval, C);

    reduce_mean_kernel<<<1, 256, 0, stream>>>(rowval, B, loss);
}